// GCN_13529146983053
// MI455X (gfx1250) — compile-verified
//
#include <hip/hip_runtime.h>

typedef __attribute__((ext_vector_type(2))) float v2f;
typedef __attribute__((ext_vector_type(8))) float v8f;

#define D 128
#define SW 132            // padded k-stride for transposed W in LDS: bank-conflict-free b64 reads
#define BN_ROWS 256

// ---------------- degree / symmetric norm ----------------
__global__ void k_deg_init(float* deg, int n) {
  int i = blockIdx.x * blockDim.x + threadIdx.x;
  if (i < n) deg[i] = 1.0f;   // self-loop contributes 1 to every node
}

__global__ void k_deg_accum(const int* __restrict__ dst, float* deg, int E) {
  int e = blockIdx.x * blockDim.x + threadIdx.x;
  if (e < E) atomicAdd(&deg[dst[e]], 1.0f);
}

__global__ void k_dinv(const float* __restrict__ deg, float* __restrict__ dinv, int n) {
  int i = blockIdx.x * blockDim.x + threadIdx.x;
  if (i < n) dinv[i] = rsqrtf(deg[i]);   // deg >= 1 always (self loops)
}

// ---------------- f32 WMMA GEMM: C[M,128] = A[M,128] @ W[128,128] ----------------
// One wave computes a 16-row strip across all 128 output columns.
// V_WMMA_F32_16X16X4_F32: A 16x4 (2 VGPR/lane), B 4x16 (2 VGPR/lane), C/D 16x16 (8 VGPR/lane).
// W is transposed into LDS (sWt[n*SW + k], SW=132) so each lane's B fragment
// {W[kb][nn], W[kb+1][nn]} is one aligned ds_load_b64 with no bank conflicts
// and no register re-pairing movs.
// NOTE: A and C may alias (in-place use) -> no __restrict__ on them; all A loads
// precede C stores in program order within the owning wave.
__global__ void __launch_bounds__(256)
k_gemm_wmma(const float* A, const float* __restrict__ W, float* C, int M) {
  __shared__ float sWt[D * SW];                     // 66 KB of the 320 KB/WGP LDS
  int tid = threadIdx.x;
  for (int i = tid; i < D * D / 4; i += 256) {      // coalesced b128 reads, transpose on store
    float4 w = ((const float4*)W)[i];               // = W[k][n0..n0+3]
    int k  = i >> 5;
    int n0 = (i & 31) << 2;
    sWt[(n0 + 0) * SW + k] = w.x;
    sWt[(n0 + 1) * SW + k] = w.y;
    sWt[(n0 + 2) * SW + k] = w.z;
    sWt[(n0 + 3) * SW + k] = w.w;
  }
  __syncthreads();

  int wave = tid >> 5;
  int lane = tid & 31;
  int row0 = (blockIdx.x * 8 + wave) * 16;
  if (row0 >= M) return;                            // whole-wave exit: EXEC stays all-1s for WMMA

  int lm = lane & 15;                               // M index within tile
  int kh = lane >> 4;                               // 0 -> K=k0,k0+1 ; 1 -> K=k0+2,k0+3

  v8f acc[8] = {};                                  // 8 N-tiles of 16 cols each

  for (int k0 = 0; k0 < D; k0 += 4) {
    int kb = k0 + kh * 2;
    v2f a = *(const v2f*)(A + (size_t)(row0 + lm) * D + kb);   // A[row][kb..kb+1]
    const float* bp = &sWt[lm * SW + kb];           // per-lane base; t adds an immediate offset
#pragma unroll
    for (int t = 0; t < 8; ++t) {
      v2f b = *(const v2f*)(bp + t * 16 * SW);      // one ds_load_b64, conflict-free
      acc[t] = __builtin_amdgcn_wmma_f32_16x16x4_f32(
          false, a, false, b, (short)0, acc[t], false, false);
    }
  }

#pragma unroll
  for (int t = 0; t < 8; ++t) {
#pragma unroll
    for (int g = 0; g < 8; ++g) {                   // C/D: VGPR g -> M=g (lanes 0-15), M=g+8 (lanes 16-31)
      C[(size_t)(row0 + kh * 8 + g) * D + (t * 16 + lm)] = acc[t][g];
    }
  }
}

// ---------------- aggregation ----------------
// agg[i,c] = bias[c] + h[i,c] * dinv[i]^2   (self-loop term + bias, full init of agg)
__global__ void k_agg_init(const float* __restrict__ h, const float* __restrict__ dinv,
                           const float* __restrict__ bias, float* __restrict__ agg, int n) {
  int idx = blockIdx.x * blockDim.x + threadIdx.x;
  if (idx < n * D) {
    int i = idx >> 7;
    int c = idx & (D - 1);
    float dv = dinv[i];
    agg[idx] = bias[c] + h[idx] * dv * dv;
  }
}

// one wave per edge; each lane moves a float4 (32 lanes * 4 = 128 features)
__global__ void k_edge_scatter(const float* __restrict__ h, const int* __restrict__ src,
                               const int* __restrict__ dst, const float* __restrict__ dinv,
                               float* agg, int E) {
  int gt = blockIdx.x * blockDim.x + threadIdx.x;
  int e = gt >> 5;
  int lane = gt & 31;
  if (e >= E) return;
  int s = src[e];
  int d = dst[e];
  float nrm = dinv[s] * dinv[d];
  float4 hv = ((const float4*)(h + (size_t)s * D))[lane];
  float* op = agg + (size_t)d * D + lane * 4;
  atomicAdd(op + 0, hv.x * nrm);
  atomicAdd(op + 1, hv.y * nrm);
  atomicAdd(op + 2, hv.z * nrm);
  atomicAdd(op + 3, hv.w * nrm);
}

// ---------------- batchnorm (training stats) + relu ----------------
__global__ void k_zero(float* p, int n) {
  int i = blockIdx.x * blockDim.x + threadIdx.x;
  if (i < n) p[i] = 0.0f;
}

__global__ void k_bn_partial(const float* __restrict__ h, float* colsum, float* colsq, int n) {
  int c = threadIdx.x;                              // 128 threads: one column each (coalesced)
  int r0 = blockIdx.x * BN_ROWS;
  int rend = r0 + BN_ROWS; if (rend > n) rend = n;
  float s = 0.0f, q = 0.0f;
  for (int r = r0; r < rend; ++r) {
    float v = h[(size_t)r * D + c];
    s += v;
    q += v * v;
  }
  atomicAdd(&colsum[c], s);
  atomicAdd(&colsq[c], q);
}

__global__ void k_bn_finalize(const float* __restrict__ colsum, const float* __restrict__ colsq,
                              const float* __restrict__ gamma, const float* __restrict__ beta,
                              float* __restrict__ bnA, float* __restrict__ bnB, float invN) {
  int c = threadIdx.x;                              // 128 threads
  float mu  = colsum[c] * invN;
  float var = colsq[c] * invN - mu * mu;            // biased variance == mean((h-mu)^2)
  float s = gamma[c] * rsqrtf(var + 1e-5f);
  bnA[c] = s;
  bnB[c] = beta[c] - mu * s;
}

__global__ void k_bn_relu(const float* __restrict__ h, const float* __restrict__ bnA,
                          const float* __restrict__ bnB, float* __restrict__ o, int total) {
  int idx = blockIdx.x * blockDim.x + threadIdx.x;
  if (idx < total) {
    int c = idx & (D - 1);
    o[idx] = fmaxf(fmaf(h[idx], bnA[c], bnB[c]), 0.0f);
  }
}

// ---------------- host ----------------
extern "C" void kernel_launch(void* const* d_in, const int* in_sizes, int n_in,
                              void* d_out, int out_size, void* d_ws, size_t ws_size,
                              hipStream_t stream) {
  const float* x     = (const float*)d_in[0];
  const int*   ei    = (const int*)  d_in[1];
  const float* W1    = (const float*)d_in[2];
  const float* b1    = (const float*)d_in[3];
  const float* gamma = (const float*)d_in[4];
  const float* beta  = (const float*)d_in[5];
  const float* W2    = (const float*)d_in[6];
  const float* b2    = (const float*)d_in[7];
  float* out = (float*)d_out;

  const int n = in_sizes[0] / D;      // 50000
  const int E = in_sizes[1] / 2;      // 800000
  const int* src = ei;                // edge_index[0]
  const int* dst = ei + E;            // edge_index[1]

  // workspace carve-up (~26 MB): deg, dinv, bn scratch, one N x D activation buffer
  float* p = (float*)d_ws;
  float* deg    = p; p += n;
  float* dinv   = p; p += n;
  float* colsum = p; p += D;
  float* colsq  = p; p += D;
  float* bnA    = p; p += D;
  float* bnB    = p; p += D;
  float* buf    = p;                  // n * D floats

  const int total = n * D;
  dim3 b256(256);
  const int gemm_blocks = ((n / 16) + 7) / 8;       // 8 waves/block, 16 rows/wave

  // normalization coefficients
  k_deg_init <<<(n + 255) / 256, b256, 0, stream>>>(deg, n);
  k_deg_accum<<<(E + 255) / 256, b256, 0, stream>>>(dst, deg, E);
  k_dinv     <<<(n + 255) / 256, b256, 0, stream>>>(deg, dinv, n);

  // layer 1: h1 = x @ W1 ; agg1 = scatter(norm * h1) + b1   (agg1 lives in d_out)
  k_gemm_wmma<<<gemm_blocks, b256, 0, stream>>>(x, W1, buf, n);
  k_agg_init <<<(total + 255) / 256, b256, 0, stream>>>(buf, dinv, b1, out, n);
  k_edge_scatter<<<(E * 32 + 255) / 256, b256, 0, stream>>>(buf, src, dst, dinv, out, E);

  // batchnorm (batch stats) + relu -> buf = h2
  k_zero<<<1, b256, 0, stream>>>(colsum, 2 * D);    // colsum & colsq are contiguous
  k_bn_partial <<<(n + BN_ROWS - 1) / BN_ROWS, dim3(D), 0, stream>>>(out, colsum, colsq, n);
  k_bn_finalize<<<1, dim3(D), 0, stream>>>(colsum, colsq, gamma, beta, bnA, bnB, 1.0f / (float)n);
  k_bn_relu    <<<(total + 255) / 256, b256, 0, stream>>>(out, bnA, bnB, buf, total);

  // layer 2: h3 = h2 @ W2 (in-place, wave-row-disjoint so safe) ; agg2 -> d_out
  k_gemm_wmma<<<gemm_blocks, b256, 0, stream>>>(buf, W2, buf, n);
  k_agg_init <<<(total + 255) / 256, b256, 0, stream>>>(buf, dinv, b2, out, n);
  k_edge_scatter<<<(E * 32 + 255) / 256, b256, 0, stream>>>(buf, src, dst, dinv, out, E);
}